// RelationConv_45174466019827
// MI455X (gfx1250) — compile-verified
//
#include <hip/hip_runtime.h>
#include <hip/hip_bf16.h>
#include <math.h>

// RelationConv on MI455X (gfx1250, wave32).
// Memory-bound scatter/gather pipeline; working set (~60MB) is L2-resident (192MB L2).
// N=50000 nodes, E=800000 edges, D=64 features (float32).
// No dense-GEMM structure -> WMMA inapplicable; the rewarded CDNA5 paths are
// wave32 lane ops and native non-returning f32 atomics serviced by L2.

#define D_FEAT 64
#define EPS_NORM 1e-12f
#define EPS_SM 1e-16f

// Native, non-returning f32 atomic add (global_atomic_add_f32, agent scope).
// Avoids any CAS-loop lowering of plain atomicAdd(float*).
__device__ __forceinline__ void atomicAddF32(float* p, float v) {
    __hip_atomic_fetch_add(p, v, __ATOMIC_RELAXED, __HIP_MEMORY_SCOPE_AGENT);
}

// Float atomic max via monotone integer mapping (native global_atomic_max_i32 / min_u32).
__device__ __forceinline__ void atomicMaxF32(float* addr, float val) {
    if (!(val < 0.0f)) {
        atomicMax((int*)addr, __float_as_int(val));
    } else {
        atomicMin((unsigned int*)addr, __float_as_uint(val));
    }
}

// K0: init per-node scratch. amax seeded with the self-loop alpha = beta*1.
__global__ void k0_init(float* __restrict__ sq, float* __restrict__ amax,
                        float* __restrict__ denom, const float* __restrict__ beta, int N) {
    int i = blockIdx.x * blockDim.x + threadIdx.x;
    if (i < N) {
        sq[i]    = 0.0f;
        denom[i] = 0.0f;
        amax[i]  = beta[0];
    }
}

// K1: sq[row] += w^2   (segment_sum of edge_attr^2 by source row)
__global__ void k1_edge_sq(const int* __restrict__ row, const float* __restrict__ ea,
                           float* __restrict__ sq, int E) {
    int e = blockIdx.x * blockDim.x + threadIdx.x;
    if (e < E) {
        float w = ea[e];
        atomicAddF32(&sq[row[e]], w * w);
    }
}

// K2: row-wise L2 normalize x -> xn. One wave32 per node, float2 per lane (64 floats).
__global__ void k2_norm_x(const float* __restrict__ x, float* __restrict__ xn, int N) {
    int t = blockIdx.x * blockDim.x + threadIdx.x;
    int node = t >> 5;
    int lane = t & 31;
    if (node >= N) return;
    const float2* xv = (const float2*)x;
    float2 v = xv[node * 32 + lane];
    float s = v.x * v.x + v.y * v.y;
    #pragma unroll
    for (int off = 16; off >= 1; off >>= 1)      // wave32 butterfly reduction
        s += __shfl_xor(s, off, 32);
    float inv = 1.0f / fmaxf(sqrtf(s), EPS_NORM);
    float2* xnv = (float2*)xn;
    xnv[node * 32 + lane] = make_float2(v.x * inv, v.y * inv);
}

// K3: normalized edge weight, scatter-max of alpha = beta*ea into amax[row]
__global__ void k3_edge_amax(const int* __restrict__ row, const float* __restrict__ eattr,
                             const float* __restrict__ sq, float* __restrict__ amax,
                             const float* __restrict__ beta, int E) {
    int e = blockIdx.x * blockDim.x + threadIdx.x;
    if (e < E) {
        int r = row[e];
        float ea = eattr[e] / fmaxf(sqrtf(sq[r]), EPS_NORM);
        atomicMaxF32(&amax[r], beta[0] * ea);
    }
}

// K4: ex = exp(beta*ea - amax[row]); stash ex; denom[row] += ex
__global__ void k4_edge_exp(const int* __restrict__ row, const float* __restrict__ eattr,
                            const float* __restrict__ sq, const float* __restrict__ amax,
                            float* __restrict__ ex, float* __restrict__ denom,
                            const float* __restrict__ beta, int E) {
    int e = blockIdx.x * blockDim.x + threadIdx.x;
    if (e < E) {
        int r = row[e];
        float ea = eattr[e] / fmaxf(sqrtf(sq[r]), EPS_NORM);
        float v  = expf(beta[0] * ea - amax[r]);
        ex[e] = v;
        atomicAddF32(&denom[r], v);
    }
}

// K5: fold self-loop into denom; out = (1+eps + a_self) * xn. One wave32 per node.
__global__ void k5_out_init(const float* __restrict__ xn, const float* __restrict__ amax,
                            float* __restrict__ denom, float* __restrict__ out,
                            const float* __restrict__ beta, const float* __restrict__ epsp,
                            int N) {
    int t = blockIdx.x * blockDim.x + threadIdx.x;
    int node = t >> 5;
    int lane = t & 31;
    if (node >= N) return;
    float am      = amax[node];
    float ex_self = expf(beta[0] - am);           // self-loop ea == 1
    float dtot    = denom[node] + ex_self;        // full softmax denominator
    if (lane == 0) denom[node] = dtot;            // store total for K6
    float coef = (1.0f + epsp[0]) + ex_self / (dtot + EPS_SM);
    const float2* xnv = (const float2*)xn;
    float2 v = xnv[node * 32 + lane];
    float2* ov = (float2*)out;
    ov[node * 32 + lane] = make_float2(coef * v.x, coef * v.y);
}

// K6: SpMM scatter: out[row] += (ex/(dtot+1e-16)) * xn[col].
// 16 lanes per edge, float4 gathers (global_load_b128), 4 f32 atomics per lane.
__global__ void k6_spmm(const int* __restrict__ row, const int* __restrict__ col,
                        const float* __restrict__ ex, const float* __restrict__ denom,
                        const float* __restrict__ xn, float* __restrict__ out, int E) {
    int t = blockIdx.x * blockDim.x + threadIdx.x;
    int e    = t >> 4;          // 16 lanes per edge
    int lane = t & 15;
    if (e >= E) return;
    int r = row[e];
    int c = col[e];
    float scale = ex[e] / (denom[r] + EPS_SM);
    const float4* xnv = (const float4*)xn;
    float4 v = xnv[c * 16 + lane];               // gather full 256B row per edge (L2-resident)
    float* dst = out + (size_t)r * D_FEAT + lane * 4;
    atomicAddF32(dst + 0, scale * v.x);          // global_atomic_add_f32 -> L2 atomic units
    atomicAddF32(dst + 1, scale * v.y);
    atomicAddF32(dst + 2, scale * v.z);
    atomicAddF32(dst + 3, scale * v.w);
}

extern "C" void kernel_launch(void* const* d_in, const int* in_sizes, int n_in,
                              void* d_out, int out_size, void* d_ws, size_t ws_size,
                              hipStream_t stream) {
    const float* x      = (const float*)d_in[0];
    const float* eattr  = (const float*)d_in[1];
    const float* beta   = (const float*)d_in[2];
    const float* epsp   = (const float*)d_in[3];
    const int*   eidx   = (const int*)d_in[4];   // shape (2, E): row then col

    const int E = in_sizes[1];
    const int N = in_sizes[0] / D_FEAT;
    const int* row = eidx;
    const int* col = eidx + E;

    // workspace layout (floats)
    float* ws    = (float*)d_ws;
    float* xn    = ws;                        // N * 64
    float* sq    = xn + (size_t)N * D_FEAT;   // N
    float* amax  = sq + N;                    // N
    float* denom = amax + N;                  // N
    float* ex    = denom + N;                 // E

    float* out = (float*)d_out;

    const int B = 256;
    const int nodeBlocks     = (N + B - 1) / B;
    const int edgeBlocks     = (E + B - 1) / B;
    const int nodeWaveBlocks = (int)(((size_t)N * 32 + B - 1) / B);  // wave32 per node
    const int edgeQuadBlocks = (int)(((size_t)E * 16 + B - 1) / B);  // 16 lanes per edge

    k0_init     <<<nodeBlocks,     B, 0, stream>>>(sq, amax, denom, beta, N);
    k1_edge_sq  <<<edgeBlocks,     B, 0, stream>>>(row, eattr, sq, E);
    k2_norm_x   <<<nodeWaveBlocks, B, 0, stream>>>(x, xn, N);
    k3_edge_amax<<<edgeBlocks,     B, 0, stream>>>(row, eattr, sq, amax, beta, E);
    k4_edge_exp <<<edgeBlocks,     B, 0, stream>>>(row, eattr, sq, amax, ex, denom, beta, E);
    k5_out_init <<<nodeWaveBlocks, B, 0, stream>>>(xn, amax, denom, out, beta, epsp, N);
    k6_spmm     <<<edgeQuadBlocks, B, 0, stream>>>(row, col, ex, denom, xn, out, E);
}